// EPMoE_50483045597482
// MI455X (gfx1250) — compile-verified
//
#include <hip/hip_runtime.h>
#include <hip/hip_bf16.h>

// ---------------------------------------------------------------------------
// MoE forward (T=512, E=8, top-2, H=I=1024, fp32) for gfx1250 using
// V_WMMA_F32_16X16X4_F32. Routed grouped-GEMM pipeline:
//   route -> gemm1(+SiLU*mul) -> gemm2(+scale*routing weight) -> combine
// M-register-blocked: each block covers 32 pair-rows so every global B
// fragment feeds 2 WMMAs (halves weight traffic vs 16-row tiles).
// ---------------------------------------------------------------------------

typedef __attribute__((ext_vector_type(2))) float v2f;
typedef __attribute__((ext_vector_type(4))) float v4f;
typedef __attribute__((ext_vector_type(8))) float v8f;

#define NUM_TOKENS 512
#define NUM_EXPERTS 8
#define TOPK 2
#define HID 1024
#define INTER 1024
#define TWO_I 2048
#define NPAIRS (NUM_TOKENS * TOPK)

#define RB 32            // pair-rows per block (2 WMMA M-subtiles)
#define KC 256           // K-chunk staged in LDS
#define NKC (HID / KC)   // 4 chunks

#define WMMA_F32(a, b, c) \
  __builtin_amdgcn_wmma_f32_16x16x4_f32(false, (a), false, (b), (short)0, (c), false, false)

// ---------------------------------------------------------------------------
// Kernel 1: routing. One block of 512 threads (one per token).
// ---------------------------------------------------------------------------
__global__ __launch_bounds__(NUM_TOKENS) void moe_route(
    const float* __restrict__ logits,   // [T, E]
    float* __restrict__ topk_w,         // [T*K]
    int* __restrict__ offs,             // [E+1]
    int* __restrict__ pair_token,       // [T*K]
    int* __restrict__ pair_tk)          // [T*K]
{
  __shared__ int cnt[NUM_EXPERTS];
  __shared__ int cur[NUM_EXPERTS];
  const int t = threadIdx.x;
  if (t < NUM_EXPERTS) cnt[t] = 0;
  __syncthreads();

  float l[NUM_EXPERTS];
#pragma unroll
  for (int e = 0; e < NUM_EXPERTS; ++e) l[e] = logits[t * NUM_EXPERTS + e];

  int e1 = 0;
#pragma unroll
  for (int e = 1; e < NUM_EXPERTS; ++e) if (l[e] > l[e1]) e1 = e;
  int e2 = -1;
#pragma unroll
  for (int e = 0; e < NUM_EXPERTS; ++e)
    if (e != e1 && (e2 < 0 || l[e] > l[e2])) e2 = e;

  float p2 = __expf(l[e2] - l[e1]);
  float inv = 1.0f / (1.0f + p2);
  topk_w[t * TOPK + 0] = inv;
  topk_w[t * TOPK + 1] = p2 * inv;

  atomicAdd(&cnt[e1], 1);
  atomicAdd(&cnt[e2], 1);
  __syncthreads();

  if (t == 0) {
    int acc = 0;
#pragma unroll
    for (int e = 0; e < NUM_EXPERTS; ++e) {
      offs[e] = acc;
      cur[e] = acc;
      acc += cnt[e];
    }
    offs[NUM_EXPERTS] = acc;
  }
  __syncthreads();

  int p1 = atomicAdd(&cur[e1], 1);
  pair_token[p1] = t;
  pair_tk[p1] = t * TOPK + 0;
  int p2i = atomicAdd(&cur[e2], 1);
  pair_token[p2i] = t;
  pair_tk[p2i] = t * TOPK + 1;
}

// ---------------------------------------------------------------------------
// Kernel 2: grouped GEMM1 + SiLU*mul.
// grid = (32 m-tiles, 8 experts), block = 1024 threads (32 waves).
// Block: 32 pair-rows x 2048 cols of X @ W13_e^T.
// Wave w owns gate/up tile-pairs i = {2w, 2w+1} and both M-subtiles.
// ---------------------------------------------------------------------------
__global__ __launch_bounds__(1024) void moe_gemm1(
    const float* __restrict__ X,        // [T, H]
    const float* __restrict__ W13,      // [E, 2I, H]
    const float* __restrict__ s13i,     // [E]
    const float* __restrict__ s13w,     // [E]
    const int* __restrict__ offs,       // [E+1]
    const int* __restrict__ pair_token, // [T*K]
    float* __restrict__ act)            // [T*K, INTER]
{
  __shared__ float As[RB][KC + 8];
  __shared__ int rowtok[RB];

  const int e = blockIdx.y;
  const int mt = blockIdx.x;
  const int off = offs[e];
  const int cnt = offs[e + 1] - off;
  if (mt * RB >= cnt) return;

  const int tid = threadIdx.x;
  if (tid < RB) {
    int gr = mt * RB + tid;
    rowtok[tid] = (gr < cnt) ? pair_token[off + gr] : -1;
  }
  __syncthreads();

  const int wave = tid >> 5;         // 0..31
  const int lane = tid & 31;
  const int lm = lane & 15;
  const int khl = (lane >> 4) * 2;
  const int sr = tid >> 5;           // staging row 0..31
  const int sc = (tid & 31) * 8;     // staging col base (8 floats/thread)

  const float scale = s13i[e] * s13w[e];

  v8f zero = {};
  v8f accG[2][2], accU[2][2];        // [tile-pair i][m-subtile]
#pragma unroll
  for (int i = 0; i < 2; ++i)
#pragma unroll
    for (int m = 0; m < 2; ++m) { accG[i][m] = zero; accU[i][m] = zero; }

  const int n0a = (wave * 2 + 0) * 16;
  const int n0b = (wave * 2 + 1) * 16;

  for (int chunk = 0; chunk < NKC; ++chunk) {
    // ---- stage 32 x 256 tile of gathered X rows into LDS ----
    {
      int tok = rowtok[sr];
      if (tok >= 0) {
        const float* src = X + (size_t)tok * HID + chunk * KC + sc;
        *(v4f*)&As[sr][sc + 0] = *(const v4f*)(src + 0);
        *(v4f*)&As[sr][sc + 4] = *(const v4f*)(src + 4);
      } else {
        v4f z4 = {};
        *(v4f*)&As[sr][sc + 0] = z4;
        *(v4f*)&As[sr][sc + 4] = z4;
      }
    }
    __syncthreads();

    const float* pg0 = W13 + ((size_t)e * TWO_I + n0a + lm) * HID + chunk * KC + khl;
    const float* pu0 = pg0 + (size_t)INTER * HID;
    const float* pg1 = W13 + ((size_t)e * TWO_I + n0b + lm) * HID + chunk * KC + khl;
    const float* pu1 = pg1 + (size_t)INTER * HID;
    const float* pa0 = &As[lm][khl];
    const float* pa1 = &As[16 + lm][khl];
    __builtin_prefetch(pg0, 0, 1);
    __builtin_prefetch(pu0, 0, 1);
    __builtin_prefetch(pg1, 0, 1);
    __builtin_prefetch(pu1, 0, 1);

#pragma unroll 4
    for (int kb = 0; kb < KC / 4; ++kb) {
      v2f a0 = *(const v2f*)(pa0 + kb * 4);
      v2f a1 = *(const v2f*)(pa1 + kb * 4);
      v2f bg0 = *(const v2f*)(pg0 + kb * 4);
      v2f bu0 = *(const v2f*)(pu0 + kb * 4);
      v2f bg1 = *(const v2f*)(pg1 + kb * 4);
      v2f bu1 = *(const v2f*)(pu1 + kb * 4);
      accG[0][0] = WMMA_F32(a0, bg0, accG[0][0]);
      accG[0][1] = WMMA_F32(a1, bg0, accG[0][1]);
      accU[0][0] = WMMA_F32(a0, bu0, accU[0][0]);
      accU[0][1] = WMMA_F32(a1, bu0, accU[0][1]);
      accG[1][0] = WMMA_F32(a0, bg1, accG[1][0]);
      accG[1][1] = WMMA_F32(a1, bg1, accG[1][1]);
      accU[1][0] = WMMA_F32(a0, bu1, accU[1][0]);
      accU[1][1] = WMMA_F32(a1, bu1, accU[1][1]);
    }
    __syncthreads();
  }

  // ---- epilogue: scale, SiLU(gate)*up, scatter to act ----
  const int mbase = (lane >> 4) * 8;   // C/D layout: M = r + 8*(lane/16)
#pragma unroll
  for (int i = 0; i < 2; ++i) {
    const int col = (wave * 2 + i) * 16 + lm;
#pragma unroll
    for (int m = 0; m < 2; ++m) {
#pragma unroll
      for (int r = 0; r < 8; ++r) {
        int gr = mt * RB + m * 16 + mbase + r;
        if (gr < cnt) {
          float g = accG[i][m][r] * scale;
          float u = accU[i][m][r] * scale;
          float a = g / (1.0f + __expf(-g)) * u;   // silu(g) * u
          act[(size_t)(off + gr) * INTER + col] = a;
        }
      }
    }
  }
}

// ---------------------------------------------------------------------------
// Kernel 3: grouped GEMM2 (act @ W2_e^T) with fused s2 * routing weight.
// grid = (32, 8), block = 1024 threads (32 waves).
// Wave w owns N-tiles {2w, 2w+1} of 64, and both M-subtiles.
// ---------------------------------------------------------------------------
__global__ __launch_bounds__(1024) void moe_gemm2(
    const float* __restrict__ act,      // [T*K, INTER]
    const float* __restrict__ W2,       // [E, H, INTER]
    const float* __restrict__ s2i,      // [E]
    const float* __restrict__ s2w,      // [E]
    const int* __restrict__ offs,       // [E+1]
    const int* __restrict__ pair_tk,    // [T*K]
    const float* __restrict__ topk_w,   // [T*K]
    float* __restrict__ pout)           // [T*K, H] slot-indexed
{
  __shared__ float As[RB][KC + 8];
  __shared__ float wrow[RB];
  __shared__ int rowdst[RB];

  const int e = blockIdx.y;
  const int mt = blockIdx.x;
  const int off = offs[e];
  const int cnt = offs[e + 1] - off;
  if (mt * RB >= cnt) return;

  const int tid = threadIdx.x;
  if (tid < RB) {
    int gr = mt * RB + tid;
    if (gr < cnt) {
      int tk = pair_tk[off + gr];
      rowdst[tid] = tk;
      wrow[tid] = topk_w[tk];
    } else {
      rowdst[tid] = -1;
      wrow[tid] = 0.0f;
    }
  }
  __syncthreads();

  const int wave = tid >> 5;
  const int lane = tid & 31;
  const int lm = lane & 15;
  const int khl = (lane >> 4) * 2;
  const int sr = tid >> 5;
  const int sc = (tid & 31) * 8;

  const float scale = s2i[e] * s2w[e];

  v8f zero = {};
  v8f acc[2][2];                       // [n-tile i][m-subtile]
#pragma unroll
  for (int i = 0; i < 2; ++i)
#pragma unroll
    for (int m = 0; m < 2; ++m) acc[i][m] = zero;

  const int n0a = (wave * 2 + 0) * 16;
  const int n0b = (wave * 2 + 1) * 16;

  for (int chunk = 0; chunk < NKC; ++chunk) {
    {
      bool valid = (mt * RB + sr) < cnt;
      if (valid) {
        const float* src = act + (size_t)(off + mt * RB + sr) * INTER + chunk * KC + sc;
        *(v4f*)&As[sr][sc + 0] = *(const v4f*)(src + 0);
        *(v4f*)&As[sr][sc + 4] = *(const v4f*)(src + 4);
      } else {
        v4f z4 = {};
        *(v4f*)&As[sr][sc + 0] = z4;
        *(v4f*)&As[sr][sc + 4] = z4;
      }
    }
    __syncthreads();

    const float* pb0 = W2 + ((size_t)e * HID + n0a + lm) * INTER + chunk * KC + khl;
    const float* pb1 = W2 + ((size_t)e * HID + n0b + lm) * INTER + chunk * KC + khl;
    const float* pa0 = &As[lm][khl];
    const float* pa1 = &As[16 + lm][khl];
    __builtin_prefetch(pb0, 0, 1);
    __builtin_prefetch(pb1, 0, 1);

#pragma unroll 4
    for (int kb = 0; kb < KC / 4; ++kb) {
      v2f a0 = *(const v2f*)(pa0 + kb * 4);
      v2f a1 = *(const v2f*)(pa1 + kb * 4);
      v2f b0 = *(const v2f*)(pb0 + kb * 4);
      v2f b1 = *(const v2f*)(pb1 + kb * 4);
      acc[0][0] = WMMA_F32(a0, b0, acc[0][0]);
      acc[0][1] = WMMA_F32(a1, b0, acc[0][1]);
      acc[1][0] = WMMA_F32(a0, b1, acc[1][0]);
      acc[1][1] = WMMA_F32(a1, b1, acc[1][1]);
    }
    __syncthreads();
  }

  const int mbase = (lane >> 4) * 8;
#pragma unroll
  for (int i = 0; i < 2; ++i) {
    const int col = (wave * 2 + i) * 16 + lm;
#pragma unroll
    for (int m = 0; m < 2; ++m) {
#pragma unroll
      for (int r = 0; r < 8; ++r) {
        int M = m * 16 + mbase + r;
        int dst = rowdst[M];
        if (dst >= 0) {
          float v = acc[i][m][r] * scale * wrow[M];
          pout[(size_t)dst * HID + col] = v;
        }
      }
    }
  }
}

// ---------------------------------------------------------------------------
// Kernel 4: combine the two slots per token (deterministic, no atomics).
// ---------------------------------------------------------------------------
__global__ __launch_bounds__(256) void moe_combine(
    const float* __restrict__ pout,   // [T*K, H]
    float* __restrict__ out)          // [T, H]
{
  int idx = blockIdx.x * 256 + threadIdx.x;   // t*H + h
  int t = idx >> 10;
  int h = idx & (HID - 1);
  out[idx] = pout[(size_t)(t * TOPK + 0) * HID + h] +
             pout[(size_t)(t * TOPK + 1) * HID + h];
}

// ---------------------------------------------------------------------------
// Host launcher
// ---------------------------------------------------------------------------
extern "C" void kernel_launch(void* const* d_in, const int* in_sizes, int n_in,
                              void* d_out, int out_size, void* d_ws, size_t ws_size,
                              hipStream_t stream) {
  const float* X    = (const float*)d_in[0];   // hidden_states [T,H]
  const float* RL   = (const float*)d_in[1];   // router_logits [T,E]
  const float* W13  = (const float*)d_in[2];   // [E, 2I, H]
  const float* W2   = (const float*)d_in[3];   // [E, H, I]
  const float* s13i = (const float*)d_in[4];
  const float* s13w = (const float*)d_in[5];
  const float* s2i  = (const float*)d_in[6];
  const float* s2w  = (const float*)d_in[7];
  float* out = (float*)d_out;

  char* ws = (char*)d_ws;
  float* topk_w     = (float*)(ws + 0);                        // 4 KB
  int*   offs       = (int*)  (ws + 4096);                     // 64 B
  int*   pair_token = (int*)  (ws + 4096 + 64);                // 4 KB
  int*   pair_tk    = (int*)  (ws + 4096 + 64 + 4096);         // 4 KB
  float* act        = (float*)(ws + 32768);                    // 4 MB
  float* pout       = (float*)(ws + 32768 + (size_t)NPAIRS * INTER * 4); // 4 MB

  moe_route<<<1, NUM_TOKENS, 0, stream>>>(RL, topk_w, offs, pair_token, pair_tk);

  dim3 grid_g(NPAIRS / RB, NUM_EXPERTS);   // (32, 8)
  moe_gemm1<<<grid_g, 1024, 0, stream>>>(X, W13, s13i, s13w, offs, pair_token, act);
  moe_gemm2<<<grid_g, 1024, 0, stream>>>(act, W2, s2i, s2w, offs, pair_tk, topk_w, pout);

  moe_combine<<<(NUM_TOKENS * HID) / 256, 256, 0, stream>>>(pout, out);
}